// PointNetSAModule_70153995813276
// MI455X (gfx1250) — compile-verified
//
#include <hip/hip_runtime.h>
#include <math.h>
#include <stdint.h>

typedef __attribute__((ext_vector_type(2))) float v2f;
typedef __attribute__((ext_vector_type(8))) float v8f;
typedef __attribute__((ext_vector_type(4))) uint32_t u32x4;
typedef __attribute__((ext_vector_type(8))) uint32_t u32x8;

constexpr int Bb = 8, Nn = 4096, Mm = 1024, Kk = 32;
constexpr int CIN = 64, CT = 64, C1 = 64, C2 = 128, Gg = 8;
constexpr float EPS = 1e-5f;
constexpr int MK = Mm * Kk; // 32768 columns per batch

// ---------------- TDM: DMA a 2-D tile (tile_d1 rows x tile_d0 elems, f32) from
// global (row stride stride_d0 elems) into LDS at lds_off, padding each LDS row
// from 64 dwords to 66 dwords (pad_interval=64dw, pad_amount=2dw) to dodge bank
// conflicts. Descriptor per CDNA5 ISA.
__device__ __forceinline__ void tdm_load_2d_f32(uint32_t lds_off, const void* gptr,
                                                uint32_t tile_d0, uint32_t tile_d1,
                                                uint32_t tensor_d0, uint32_t tensor_d1,
                                                uint64_t stride_d0) {
    uint64_t ga = (uint64_t)gptr;
    u32x4 g0;
    g0[0] = 1u;                                   // count=1, user descriptor
    g0[1] = lds_off;                              // lds_addr (bytes)
    g0[2] = (uint32_t)ga;                         // global_addr[31:0]
    g0[3] = (uint32_t)(ga >> 32) | (2u << 30);    // global_addr[56:32] | type=2
    u32x8 g1;
    g1[0] = (2u << 16)      // data_size = 2 -> 4 bytes
          | (1u << 20)      // pad_enable
          | (5u << 22)      // pad_interval code 5 -> every 64 dwords
          | (1u << 25);     // pad_amount  code 1 -> 2 dwords
    g1[1] = (tensor_d0 & 0xFFFFu) << 16;                          // tensor_dim0[15:0]  @ bits 63:48
    g1[2] = (tensor_d0 >> 16) | ((tensor_d1 & 0xFFFFu) << 16);    // dim0[31:16], dim1[15:0]
    g1[3] = (tensor_d1 >> 16) | ((tile_d0 & 0xFFFFu) << 16);      // dim1[31:16], tile_dim0
    g1[4] = (tile_d1 & 0xFFFFu);                                  // tile_dim1 (tile_dim2=0)
    g1[5] = (uint32_t)stride_d0;                                  // tensor_dim0_stride[31:0]
    g1[6] = (uint32_t)(stride_d0 >> 32) & 0xFFFFu;                // stride[47:32] (dim1_stride=0)
    g1[7] = 0u;
    asm volatile("tensor_load_to_lds %0, %1" :: "s"(g0), "s"(g1) : "memory");
}

// ---------------- transpose [B][64][N] -> [B][N][64] (coalesced gathers later)
__global__ void transpose64(const float* __restrict__ src, float* __restrict__ dst) {
    __shared__ float tile[64][65];
    int b = blockIdx.y, n0 = blockIdx.x * 64;
    int j = threadIdx.x & 63, q = threadIdx.x >> 6; // 256 threads
    for (int r = 0; r < 16; ++r) {
        int c = r * 4 + q;
        tile[c][j] = src[(b * 64 + c) * Nn + n0 + j];
    }
    __syncthreads();
    for (int r = 0; r < 16; ++r) {
        int n = r * 4 + q;
        dst[((size_t)(b * Nn + n0 + n)) * 64 + j] = tile[j][n];
    }
}

// ---------------- furthest point sampling: one block per batch, writes centers
__global__ void fps_kernel(const float* __restrict__ coords, float* __restrict__ centers) {
    __shared__ float dist[Nn];
    __shared__ float rv[256];
    __shared__ int   ri[256];
    int b = blockIdx.x, tid = threadIdx.x;
    const float* cb = coords + (size_t)b * 3 * Nn;
    float x0 = cb[0], y0 = cb[Nn], z0 = cb[2 * Nn];
    for (int i = tid; i < Nn; i += 256) {
        float dx = cb[i] - x0, dy = cb[Nn + i] - y0, dz = cb[2 * Nn + i] - z0;
        dist[i] = dx * dx + dy * dy + dz * dz;
    }
    if (tid == 0) {
        centers[(b * 3 + 0) * Mm] = x0;
        centers[(b * 3 + 1) * Mm] = y0;
        centers[(b * 3 + 2) * Mm] = z0;
    }
    __syncthreads();
    for (int m = 1; m < Mm; ++m) {
        float vmax = -1.f; int imax = 0;
        for (int i = tid; i < Nn; i += 256) {
            float d = dist[i];
            if (d > vmax) { vmax = d; imax = i; }
        }
        rv[tid] = vmax; ri[tid] = imax;
        __syncthreads();
        for (int s = 128; s > 0; s >>= 1) {
            if (tid < s) {
                float ov = rv[tid + s]; int oi = ri[tid + s];
                if (ov > rv[tid] || (ov == rv[tid] && oi < ri[tid])) { rv[tid] = ov; ri[tid] = oi; }
            }
            __syncthreads();
        }
        int far = ri[0];
        float fx = cb[far], fy = cb[Nn + far], fz = cb[2 * Nn + far];
        if (tid == 0) {
            centers[(b * 3 + 0) * Mm + m] = fx;
            centers[(b * 3 + 1) * Mm + m] = fy;
            centers[(b * 3 + 2) * Mm + m] = fz;
        }
        for (int i = tid; i < Nn; i += 256) {
            float dx = cb[i] - fx, dy = cb[Nn + i] - fy, dz = cb[2 * Nn + i] - fz;
            dist[i] = fminf(dist[i], dx * dx + dy * dy + dz * dz);
        }
        __syncthreads();
    }
}

// ---------------- kNN: one block per (batch, center), iterative argmin top-K
__global__ void knn_kernel(const float* __restrict__ coords, const float* __restrict__ centers,
                           int* __restrict__ nidx) {
    __shared__ float dist[Nn];
    __shared__ float rv[128];
    __shared__ int   ri[128];
    int b = blockIdx.y, m = blockIdx.x, tid = threadIdx.x;
    const float* cb = coords + (size_t)b * 3 * Nn;
    float cx = centers[(b * 3 + 0) * Mm + m];
    float cy = centers[(b * 3 + 1) * Mm + m];
    float cz = centers[(b * 3 + 2) * Mm + m];
    for (int i = tid; i < Nn; i += 128) {
        float dx = cb[i] - cx, dy = cb[Nn + i] - cy, dz = cb[2 * Nn + i] - cz;
        dist[i] = dx * dx + dy * dy + dz * dz;
    }
    __syncthreads();
    for (int k = 0; k < Kk; ++k) {
        float vmin = 3.0e38f; int imin = Nn;
        for (int i = tid; i < Nn; i += 128) {
            float d = dist[i];
            if (d < vmin) { vmin = d; imin = i; }
        }
        rv[tid] = vmin; ri[tid] = imin;
        __syncthreads();
        for (int s = 64; s > 0; s >>= 1) {
            if (tid < s) {
                float ov = rv[tid + s]; int oi = ri[tid + s];
                if (ov < rv[tid] || (ov == rv[tid] && oi < ri[tid])) { rv[tid] = ov; ri[tid] = oi; }
            }
            __syncthreads();
        }
        if (tid == 0) {
            nidx[((size_t)(b * Mm + m)) * Kk + k] = ri[0];
            dist[ri[0]] = 3.4e38f;
        }
        __syncthreads();
    }
}

// ---------------- temb gather + max over K (coalesced rows of tembT)
__global__ void gtemb_max_kernel(const float* __restrict__ tembT, const int* __restrict__ nidx,
                                 float* __restrict__ out) {
    int b = blockIdx.y;
    int m = blockIdx.x * blockDim.y + threadIdx.y;
    int c = threadIdx.x; // 0..63
    const int* ni = nidx + ((size_t)(b * Mm + m)) * Kk;
    float best = -3.4e38f;
    for (int k = 0; k < Kk; ++k) {
        int id = ni[k];
        best = fmaxf(best, tembT[((size_t)(b * Nn + id)) * CT + c]);
    }
    out[(b * CT + c) * Mm + m] = best;
}

// ---------------- GEMM1: H1[o,col] = W1[o,:] . X[:,col] (+b1), gather fused
__global__ __launch_bounds__(128) void gemm1_kernel(
    const float* __restrict__ W1, const float* __restrict__ b1v,
    const float* __restrict__ coords, const float* __restrict__ centers,
    const int* __restrict__ nidx, const float* __restrict__ featT,
    float* __restrict__ H1) {
    constexpr int CP = 68; // 3 + 64 padded to multiple of 4
    __shared__ float Ws[64][CP];
    __shared__ float Xs[CP][66];
    int b = blockIdx.y;
    int colbase = blockIdx.x * 64;
    int tid = threadIdx.x;
    for (int t = tid; t < 64 * CP; t += 128) {
        int r = t / CP, c = t % CP;
        Ws[r][c] = (c < 67) ? W1[r * 67 + c] : 0.f;
    }
    {
        int lane = tid & 63, jhalf = tid >> 6;
        for (int r = 0; r < 32; ++r) {
            int j = r * 2 + jhalf;
            int col = colbase + j;
            int m = col >> 5, k = col & 31;
            int id = nidx[((size_t)(b * Mm + m)) * Kk + k];
            Xs[3 + lane][j] = featT[((size_t)(b * Nn + id)) * 64 + lane];
        }
        if (tid < 64) {
            int j = tid, col = colbase + j;
            int m = col >> 5, k = col & 31;
            int id = nidx[((size_t)(b * Mm + m)) * Kk + k];
            Xs[0][j] = coords[(b * 3 + 0) * Nn + id] - centers[(b * 3 + 0) * Mm + m];
            Xs[1][j] = coords[(b * 3 + 1) * Nn + id] - centers[(b * 3 + 1) * Mm + m];
            Xs[2][j] = coords[(b * 3 + 2) * Nn + id] - centers[(b * 3 + 2) * Mm + m];
            Xs[67][j] = 0.f;
        }
    }
    __syncthreads();
    int lane = tid & 31, wave = tid >> 5;
    int n16 = lane & 15, hi = lane >> 4;
    v8f z = {0.f, 0.f, 0.f, 0.f, 0.f, 0.f, 0.f, 0.f};
    v8f acc[4];
    for (int rt = 0; rt < 4; ++rt) acc[rt] = z;
    for (int kk = 0; kk < 17; ++kk) {
        int ksel = kk * 4 + 2 * hi;
        v2f bv;
        bv.x = Xs[ksel][wave * 16 + n16];
        bv.y = Xs[ksel + 1][wave * 16 + n16];
#pragma unroll
        for (int rt = 0; rt < 4; ++rt) {
            v2f av;
            av.x = Ws[rt * 16 + n16][ksel];
            av.y = Ws[rt * 16 + n16][ksel + 1];
            acc[rt] = __builtin_amdgcn_wmma_f32_16x16x4_f32(
                false, av, false, bv, (short)0, acc[rt], false, false);
        }
    }
#pragma unroll
    for (int rt = 0; rt < 4; ++rt)
#pragma unroll
        for (int v = 0; v < 8; ++v) {
            int row = rt * 16 + v + hi * 8;
            int col = colbase + wave * 16 + n16;
            H1[((size_t)(b * 64 + row) << 15) + col] = acc[rt][v] + b1v[row];
        }
}

// ---------------- GroupNorm stats over (C/G, M, K): one block per (b, g)
__global__ void gn_stats_kernel(const float* __restrict__ H, int C,
                                float* __restrict__ mean, float* __restrict__ rstd) {
    __shared__ float ssum[256], ssq[256];
    int b = blockIdx.y, g = blockIdx.x, tid = threadIdx.x;
    int cpg = C / Gg;
    size_t base = ((size_t)(b * C + g * cpg)) << 15;
    size_t cnt = (size_t)cpg << 15;
    float s = 0.f, q = 0.f;
    for (size_t i = tid; i < cnt; i += 256) { float x = H[base + i]; s += x; q += x * x; }
    ssum[tid] = s; ssq[tid] = q;
    __syncthreads();
    for (int st = 128; st > 0; st >>= 1) {
        if (tid < st) { ssum[tid] += ssum[tid + st]; ssq[tid] += ssq[tid + st]; }
        __syncthreads();
    }
    if (tid == 0) {
        float mu = ssum[0] / (float)cnt;
        float var = ssq[0] / (float)cnt - mu * mu;
        mean[b * Gg + g] = mu;
        rstd[b * Gg + g] = rsqrtf(var + EPS);
    }
}

// ---------------- normalize + swish, in place
__global__ void norm_swish_kernel(float* __restrict__ H, int C,
                                  const float* __restrict__ mean, const float* __restrict__ rstd,
                                  const float* __restrict__ gamma, const float* __restrict__ beta) {
    size_t i = (size_t)blockIdx.x * blockDim.x + threadIdx.x;
    int bc = (int)(i >> 15);
    int c = bc % C, b = bc / C;
    int g = c / (C / Gg);
    float x = H[i];
    float y = (x - mean[b * Gg + g]) * rstd[b * Gg + g] * gamma[c] + beta[c];
    H[i] = y / (1.f + __expf(-y));
}

// ---------------- GEMM2: H2[o,col] = W2[o,:] . H1n[:,col] (+b2)
// Operand staging done by the Tensor Data Mover (one DMA per tile), closed with
// s_wait_tensorcnt; the waves issue no vector-memory staging instructions.
__global__ __launch_bounds__(128) void gemm2_kernel(
    const float* __restrict__ W2, const float* __restrict__ b2v,
    const float* __restrict__ H1, float* __restrict__ H2) {
    __shared__ float Ws[128][66];
    __shared__ float Xs[64][66];
    int b = blockIdx.y;
    int colbase = blockIdx.x * 64;
    int tid = threadIdx.x;
    if (tid == 0) {
        // W2: 128 rows x 64 cols, contiguous (row stride 64 elems)
        tdm_load_2d_f32((uint32_t)(uintptr_t)&Ws[0][0], W2,
                        /*tile_d0=*/64, /*tile_d1=*/128,
                        /*tensor_d0=*/64, /*tensor_d1=*/128, /*stride_d0=*/64);
        // X: 64 channel-rows x 64 cols slab of H1, row stride 32768 elems
        tdm_load_2d_f32((uint32_t)(uintptr_t)&Xs[0][0],
                        H1 + ((size_t)(b * 64) << 15) + colbase,
                        /*tile_d0=*/64, /*tile_d1=*/64,
                        /*tensor_d0=*/MK - colbase, /*tensor_d1=*/64, /*stride_d0=*/MK);
    }
    __builtin_amdgcn_s_wait_tensorcnt(0);
    __syncthreads();
    int lane = tid & 31, wave = tid >> 5;
    int n16 = lane & 15, hi = lane >> 4;
    v8f z = {0.f, 0.f, 0.f, 0.f, 0.f, 0.f, 0.f, 0.f};
    v8f acc[8];
    for (int rt = 0; rt < 8; ++rt) acc[rt] = z;
    for (int kk = 0; kk < 16; ++kk) {
        int ksel = kk * 4 + 2 * hi;
        v2f bv;
        bv.x = Xs[ksel][wave * 16 + n16];
        bv.y = Xs[ksel + 1][wave * 16 + n16];
#pragma unroll
        for (int rt = 0; rt < 8; ++rt) {
            v2f av;
            av.x = Ws[rt * 16 + n16][ksel];
            av.y = Ws[rt * 16 + n16][ksel + 1];
            acc[rt] = __builtin_amdgcn_wmma_f32_16x16x4_f32(
                false, av, false, bv, (short)0, acc[rt], false, false);
        }
    }
#pragma unroll
    for (int rt = 0; rt < 8; ++rt)
#pragma unroll
        for (int v = 0; v < 8; ++v) {
            int row = rt * 16 + v + hi * 8;
            H2[((size_t)(b * 128 + row) << 15) + colbase + wave * 16 + n16] = acc[rt][v] + b2v[row];
        }
}

// ---------------- fused GN2 + swish + max over K -> output
__global__ void final_max_kernel(const float* __restrict__ H2,
                                 const float* __restrict__ mean, const float* __restrict__ rstd,
                                 const float* __restrict__ gamma, const float* __restrict__ beta,
                                 float* __restrict__ out) {
    int t = blockIdx.x * blockDim.x + threadIdx.x;
    int m = t & (Mm - 1);
    int o = (t >> 10) & (C2 - 1);
    int b = t >> 17;
    int g = o / (C2 / Gg);
    float mu = mean[b * Gg + g], rs = rstd[b * Gg + g], ga = gamma[o], be = beta[o];
    size_t base = ((size_t)(b * C2 + o) << 15) + (size_t)m * Kk;
    float best = -3.4e38f;
    for (int k = 0; k < Kk; ++k) {
        float x = H2[base + k];
        float y = (x - mu) * rs * ga + be;
        float sw = y / (1.f + __expf(-y));
        best = fmaxf(best, sw);
    }
    out[((size_t)(b * C2 + o)) * Mm + m] = best;
}

extern "C" void kernel_launch(void* const* d_in, const int* in_sizes, int n_in,
                              void* d_out, int out_size, void* d_ws, size_t ws_size,
                              hipStream_t stream) {
    const float* features = (const float*)d_in[0];
    const float* coords   = (const float*)d_in[1];
    const float* temb     = (const float*)d_in[2];
    const float* W1  = (const float*)d_in[3];
    const float* b1  = (const float*)d_in[4];
    const float* g1  = (const float*)d_in[5];
    const float* be1 = (const float*)d_in[6];
    const float* W2  = (const float*)d_in[7];
    const float* b2  = (const float*)d_in[8];
    const float* g2  = (const float*)d_in[9];
    const float* be2 = (const float*)d_in[10];

    float* out     = (float*)d_out;
    float* out_h   = out;                            // [B,128,M]
    float* out_ctr = out + (size_t)Bb * C2 * Mm;     // [B,3,M]
    float* out_te  = out_ctr + (size_t)Bb * 3 * Mm;  // [B,64,M]

    char* ws = (char*)d_ws;
    int*   nidx  = (int*)  (ws + 0);             // B*M*K ints   (1 MiB)
    float* featT = (float*)(ws + (2ull  << 20)); // [B,N,64]     (8 MiB)
    float* tembT = (float*)(ws + (12ull << 20)); // [B,N,64]     (8 MiB)
    float* H1    = (float*)(ws + (24ull << 20)); // [B,64,MK]   (64 MiB)
    float* H2    = (float*)(ws + (96ull << 20)); // [B,128,MK] (128 MiB)
    float* stats = (float*)(ws + (228ull << 20));
    float* mean1 = stats, *rstd1 = stats + 64, *mean2 = stats + 128, *rstd2 = stats + 192;

    transpose64<<<dim3(Nn / 64, Bb), 256, 0, stream>>>(features, featT);
    transpose64<<<dim3(Nn / 64, Bb), 256, 0, stream>>>(temb, tembT);
    fps_kernel<<<Bb, 256, 0, stream>>>(coords, out_ctr);
    knn_kernel<<<dim3(Mm, Bb), 128, 0, stream>>>(coords, out_ctr, nidx);
    gtemb_max_kernel<<<dim3(Mm / 4, Bb), dim3(64, 4), 0, stream>>>(tembT, nidx, out_te);
    gemm1_kernel<<<dim3(MK / 64, Bb), 128, 0, stream>>>(W1, b1, coords, out_ctr, nidx, featT, H1);
    gn_stats_kernel<<<dim3(Gg, Bb), 256, 0, stream>>>(H1, C1, mean1, rstd1);
    norm_swish_kernel<<<(Bb * C1 * MK) / 256, 256, 0, stream>>>(H1, C1, mean1, rstd1, g1, be1);
    gemm2_kernel<<<dim3(MK / 64, Bb), 128, 0, stream>>>(W2, b2, H1, H2);
    gn_stats_kernel<<<dim3(Gg, Bb), 256, 0, stream>>>(H2, C2, mean2, rstd2);
    final_max_kernel<<<(Bb * C2 * Mm) / 256, 256, 0, stream>>>(H2, mean2, rstd2, g2, be2, out_h);
}